// SpatialAttention_27118423507636
// MI455X (gfx1250) — compile-verified
//
#include <hip/hip_runtime.h>
#include <hip/hip_bf16.h>

typedef float v2f __attribute__((ext_vector_type(2)));
typedef float v8f __attribute__((ext_vector_type(8)));

#define WMMA_F32_K4(a, b, c) \
    __builtin_amdgcn_wmma_f32_16x16x4_f32(false, (a), false, (b), (short)0, (c), false, false)

#define BATCH 4
#define SEQ   1024
#define HD    64

// ---------------------------------------------------------------------------
// Kernel 1: hi = x @ W1a^T ; hjb = x @ W1b^T + b1
// x: [B*N][64], W1: [64][128] (row h, cols 0..63 = W1a, 64..127 = W1b)
// One wave (32 lanes) per 16x16 output tile; K=64 -> 16 WMMA f32 16x16x4 steps.
// Tiles: 256 row-tiles x 8 col-tiles (cols 0..3 -> hi, 4..7 -> hjb).
// ---------------------------------------------------------------------------
__global__ __launch_bounds__(256) void k1_gemm_hi_hjb(
    const float* __restrict__ x,
    const float* __restrict__ W1,
    const float* __restrict__ b1,
    float* __restrict__ hi,
    float* __restrict__ hjb)
{
    const int wave  = threadIdx.x >> 5;
    const int lane  = threadIdx.x & 31;
    const int tile  = blockIdx.x * 8 + wave;   // 0..2047
    const int rowT  = tile >> 3;               // 0..255
    const int colT  = tile & 7;                // 0..7
    const int i0    = rowT * 16;
    const int m     = lane & 15;               // M row (A) / N col (B)
    const int half  = lane >> 4;               // lane half
    const int kbase = half * 2;                // K pair base within a K=4 step

    const int   hbase = (colT & 3) * 16;       // output h column base (0..48)
    const int   dofs  = (colT >= 4) ? HD : 0;  // W1a vs W1b column offset
    const float* xrow = x + (size_t)(i0 + m) * HD;
    const float* wrow = W1 + (size_t)(hbase + m) * (2 * HD) + dofs;

    v8f c = {0.f, 0.f, 0.f, 0.f, 0.f, 0.f, 0.f, 0.f};
#pragma unroll
    for (int k = 0; k < 16; ++k) {
        const int d = k * 4 + kbase;
        v2f a, b;
        a.x = xrow[d];
        a.y = xrow[d + 1];
        // B[k_local][n] = W1[hbase+n][dofs + k*4 + k_local]; lane n == m here
        b.x = wrow[d];
        b.y = wrow[d + 1];
        c = WMMA_F32_K4(a, b, c);
    }

    float*      out  = (colT >= 4) ? hjb : hi;
    const float bias = (colT >= 4) ? b1[hbase + m] : 0.0f;
#pragma unroll
    for (int r = 0; r < 8; ++r) {
        const int mm = r + half * 8;           // C layout: VGPR r, lane-half picks +8
        out[(size_t)(i0 + mm) * HD + hbase + m] = c[r] + bias;
    }
}

// ---------------------------------------------------------------------------
// Kernel 2: logits + softmax -> attn
// One workgroup (256 threads) per (batch, 16-row tile). Each thread owns 4 j
// columns (j = chunk*256 + tid). hi tile cached in LDS, hjb row in registers.
// logits kept in registers (64 floats/thread), LDS tree reductions for
// row-max and row-sum, then normalized attn written out.
// ---------------------------------------------------------------------------
__global__ __launch_bounds__(256) void k2_logits_softmax(
    const float* __restrict__ hi,
    const float* __restrict__ hjb,
    const float* __restrict__ W2,
    float* __restrict__ attn)
{
    __shared__ float s_hi[16 * HD];
    __shared__ float s_w2[HD];
    __shared__ float s_red[256];

    const int bt = blockIdx.x;       // 0..255
    const int b  = bt >> 6;          // batch
    const int it = bt & 63;          // row tile within batch
    const int i0 = it * 16;
    const int t  = threadIdx.x;

    // stage hi tile (1024 floats) and W2 into LDS
    {
        const float4* src = (const float4*)(hi + (size_t)(b * SEQ + i0) * HD);
        ((float4*)s_hi)[t] = src[t];
    }
    if (t < HD) s_w2[t] = W2[t];
    __syncthreads();

    float L[4][16];
#pragma unroll
    for (int chunk = 0; chunk < 4; ++chunk) {
        const int    j  = chunk * 256 + t;
        const float* hj = hjb + (size_t)(b * SEQ + j) * HD;
        float hr[HD];
#pragma unroll
        for (int q = 0; q < HD / 4; ++q) {
            const float4 v = ((const float4*)hj)[q];
            hr[q * 4 + 0] = v.x; hr[q * 4 + 1] = v.y;
            hr[q * 4 + 2] = v.z; hr[q * 4 + 3] = v.w;
        }
#pragma unroll
        for (int i = 0; i < 16; ++i) {
            float acc = 0.0f;
#pragma unroll
            for (int h = 0; h < HD; ++h) {
                float u = s_hi[i * HD + h] + hr[h];
                u = fmaxf(u, 0.0f);                 // relu (b1 already folded in hjb)
                acc = fmaf(u, s_w2[h], acc);
            }
            L[chunk][i] = acc;                      // b2 omitted: softmax-invariant
        }
    }

    float* attnb = attn + ((size_t)b * SEQ + i0) * SEQ;
#pragma unroll
    for (int i = 0; i < 16; ++i) {
        // row max over 1024 j
        float m = fmaxf(fmaxf(L[0][i], L[1][i]), fmaxf(L[2][i], L[3][i]));
        s_red[t] = m;
        __syncthreads();
        for (int s = 128; s > 0; s >>= 1) {
            if (t < s) s_red[t] = fmaxf(s_red[t], s_red[t + s]);
            __syncthreads();
        }
        const float rowmax = s_red[0];
        __syncthreads();

        const float e0 = __expf(L[0][i] - rowmax);
        const float e1 = __expf(L[1][i] - rowmax);
        const float e2 = __expf(L[2][i] - rowmax);
        const float e3 = __expf(L[3][i] - rowmax);

        s_red[t] = e0 + e1 + e2 + e3;
        __syncthreads();
        for (int s = 128; s > 0; s >>= 1) {
            if (t < s) s_red[t] += s_red[t + s];
            __syncthreads();
        }
        const float inv = 1.0f / s_red[0];
        __syncthreads();

        float* arow = attnb + (size_t)i * SEQ;
        arow[0 * 256 + t] = e0 * inv;
        arow[1 * 256 + t] = e1 * inv;
        arow[2 * 256 + t] = e2 * inv;
        arow[3 * 256 + t] = e3 * inv;
    }
}

// ---------------------------------------------------------------------------
// Kernel 3: updated = attn @ x   (per batch: 1024x1024 @ 1024x64)
// One wave per 16x16 output tile; K=1024 -> 256 WMMA f32 16x16x4 steps.
// Tiles: B(4) x rowTiles(64) x colTiles(4) = 1024 waves = 128 blocks x 8.
// ---------------------------------------------------------------------------
__global__ __launch_bounds__(256) void k3_updated(
    const float* __restrict__ attn,
    const float* __restrict__ x,
    float* __restrict__ upd)
{
    const int wave  = threadIdx.x >> 5;
    const int lane  = threadIdx.x & 31;
    const int tile  = blockIdx.x * 8 + wave;   // 0..1023
    const int b     = tile >> 8;               // 256 tiles per batch
    const int rt    = (tile >> 2) & 63;
    const int ct    = tile & 3;
    const int i0    = rt * 16;
    const int n0    = ct * 16;
    const int m     = lane & 15;
    const int half  = lane >> 4;
    const int kbase = half * 2;

    const float* arow = attn + ((size_t)b * SEQ + i0 + m) * SEQ;
    const float* xb   = x + (size_t)b * SEQ * HD + n0 + m;

    v8f c = {0.f, 0.f, 0.f, 0.f, 0.f, 0.f, 0.f, 0.f};
#pragma unroll 4
    for (int k = 0; k < SEQ / 4; ++k) {
        const int kk = k * 4 + kbase;
        v2f a, bb;
        a.x  = arow[kk];
        a.y  = arow[kk + 1];
        bb.x = xb[(size_t)kk * HD];
        bb.y = xb[(size_t)(kk + 1) * HD];
        c = WMMA_F32_K4(a, bb, c);
    }

#pragma unroll
    for (int r = 0; r < 8; ++r) {
        const int mm = r + half * 8;
        upd[((size_t)b * SEQ + i0 + mm) * HD + n0 + m] = c[r];
    }
}

// ---------------------------------------------------------------------------
extern "C" void kernel_launch(void* const* d_in, const int* in_sizes, int n_in,
                              void* d_out, int out_size, void* d_ws, size_t ws_size,
                              hipStream_t stream) {
    (void)in_sizes; (void)n_in; (void)out_size; (void)ws_size;

    const float* x  = (const float*)d_in[0];   // [4,1024,64]
    const float* W1 = (const float*)d_in[1];   // [64,128]
    const float* b1 = (const float*)d_in[2];   // [64]
    const float* W2 = (const float*)d_in[3];   // [1,64]
    // d_in[4] = b2: softmax-invariant shift, not needed.

    float* upd  = (float*)d_out;                                   // [4,1024,64]
    float* attn = (float*)d_out + (size_t)BATCH * SEQ * HD;        // [4,1024,1024]

    float* hi  = (float*)d_ws;                                     // [4096,64]
    float* hjb = hi + (size_t)BATCH * SEQ * HD;                    // [4096,64] (+b1)

    k1_gemm_hi_hjb<<<256, 256, 0, stream>>>(x, W1, b1, hi, hjb);
    k2_logits_softmax<<<256, 256, 0, stream>>>(hi, hjb, W2, attn);
    k3_updated<<<128, 256, 0, stream>>>(attn, x, upd);
}